// TriangleScene_63436666962257
// MI455X (gfx1250) — compile-verified
//
#include <hip/hip_runtime.h>
#include <math.h>
#include <stdint.h>

#define EPSF      1e-6f
#define HIT_TOLF  1e-3f
#define MIN_LENF  (10.0f * 1.1920928955078125e-7f)   // 10 * FLT_EPSILON

#define TMAX   2048        // static LDS table capacity (T=2000 in this benchmark)
#define BLOCK  256         // 8 wave32 waves per workgroup

struct F3 { float x, y, z; };

__device__ __forceinline__ F3 mkf3(float x, float y, float z) { F3 r; r.x = x; r.y = y; r.z = z; return r; }
__device__ __forceinline__ F3 sub3(F3 a, F3 b) { return mkf3(a.x - b.x, a.y - b.y, a.z - b.z); }
__device__ __forceinline__ F3 add3(F3 a, F3 b) { return mkf3(a.x + b.x, a.y + b.y, a.z + b.z); }
__device__ __forceinline__ F3 scl3(F3 a, float s) { return mkf3(a.x * s, a.y * s, a.z * s); }
__device__ __forceinline__ float dot3(F3 a, F3 b) { return a.x * b.x + a.y * b.y + a.z * b.z; }
__device__ __forceinline__ F3 cross3(F3 a, F3 b) {
  return mkf3(a.y * b.z - a.z * b.y, a.z * b.x - a.x * b.z, a.x * b.y - a.y * b.x);
}
__device__ __forceinline__ F3 ld3(const float* p) { return mkf3(p[0], p[1], p[2]); }

// Moller-Trumbore: matches the reference's NaN/degenerate semantics
// (comparisons with NaN are false; f forced to 0 when |a| <= eps).
__device__ __forceinline__ void mt(F3 o, F3 d, F3 v0, F3 e1, F3 e2, float& t, bool& hit) {
  F3 h = cross3(d, e2);
  float a = dot3(e1, h);
  bool ok = fabsf(a) > EPSF;
  float f = ok ? (1.0f / a) : 0.0f;
  F3 s = sub3(o, v0);
  float u = f * dot3(s, h);
  F3 q = cross3(s, e1);
  float v = f * dot3(d, q);
  t = f * dot3(e2, q);
  hit = ok && (u >= 0.0f) && (v >= 0.0f) && ((u + v) <= 1.0f);
}

// Kernel A: expand vertices[triangles] -> linear [T][9] (v0, e1, e2) table in d_ws.
__global__ void tri_prep_kernel(const float* __restrict__ verts,
                                const int*   __restrict__ tris,
                                float*       __restrict__ triData, int T) {
  int t = blockIdx.x * blockDim.x + threadIdx.x;
  if (t >= T) return;
  int i0 = tris[3 * t + 0], i1 = tris[3 * t + 1], i2 = tris[3 * t + 2];
  F3 v0 = ld3(verts + 3 * i0);
  F3 v1 = ld3(verts + 3 * i1);
  F3 v2 = ld3(verts + 3 * i2);
  float* o = triData + 9 * (size_t)t;
  o[0] = v0.x;        o[1] = v0.y;        o[2] = v0.z;
  o[3] = v1.x - v0.x; o[4] = v1.y - v0.y; o[5] = v1.z - v0.z;
  o[6] = v2.x - v0.x; o[7] = v2.y - v0.y; o[8] = v2.z - v0.z;
}

// Kernel B: one workgroup per (tx, rx, path-candidate).
__global__ __launch_bounds__(BLOCK) void trace_kernel(
    const float* __restrict__ verts,   const float* __restrict__ normals,
    const float* __restrict__ txv,     const float* __restrict__ rxv,
    const int*   __restrict__ tris,    const int*   __restrict__ pc,
    const float* __restrict__ triData, int useWs,
    float* __restrict__ outPaths, float* __restrict__ outMask, float* __restrict__ outObj,
    int T, int Pn, int nrx)
{
  __shared__ float s_tri[TMAX * 9];   // 72KB for T=2000 (<=320KB/WGP on CDNA5)
  __shared__ int   s_blocked;
  const int tid = threadIdx.x;
  if (tid == 0) s_blocked = 0;

  // ---- Stage triangle table into LDS ------------------------------------
  if (useWs) {
    // gfx1250 prefetch of this block's working set (emits global_prefetch_b8)
    __builtin_prefetch(triData + 4 * tid, 0, 0);
    const int nF   = T * 9;
    const int n16  = nF >> 2;     // float4 chunks
#if defined(__gfx1250__) && __has_builtin(__builtin_amdgcn_global_load_async_to_lds_b128)
    // CDNA5 async global->LDS DMA path (ASYNCcnt-tracked, no VGPR staging)
    typedef int v4i __attribute__((ext_vector_type(4)));
    v4i* g = (v4i*)(uintptr_t)(const void*)triData;
    v4i* l = (v4i*)(uintptr_t)(void*)&s_tri[0];
    for (int i = tid; i < n16; i += BLOCK)
      __builtin_amdgcn_global_load_async_to_lds_b128(g + i, l + i, 0, 0);
#if __has_builtin(__builtin_amdgcn_s_wait_asynccnt)
    __builtin_amdgcn_s_wait_asynccnt(0);
#else
    asm volatile("s_wait_asynccnt 0x0" ::: "memory");
#endif
#else
    const float4* src = (const float4*)triData;
    float4*       dst = (float4*)&s_tri[0];
    for (int i = tid; i < n16; i += BLOCK) dst[i] = src[i];
#endif
    for (int i = (n16 << 2) + tid; i < nF; i += BLOCK)  // tail (none when T%4==0)
      s_tri[i] = triData[i];
  } else {
    // Fallback: build the table by gathering (if d_ws was too small).
    for (int t = tid; t < T; t += BLOCK) {
      int i0 = tris[3 * t + 0], i1 = tris[3 * t + 1], i2 = tris[3 * t + 2];
      F3 v0 = ld3(verts + 3 * i0);
      F3 v1 = ld3(verts + 3 * i1);
      F3 v2 = ld3(verts + 3 * i2);
      float* o = s_tri + 9 * t;
      o[0] = v0.x;        o[1] = v0.y;        o[2] = v0.z;
      o[3] = v1.x - v0.x; o[4] = v1.y - v0.y; o[5] = v1.z - v0.z;
      o[6] = v2.x - v0.x; o[7] = v2.y - v0.y; o[8] = v2.z - v0.z;
    }
  }
  __syncthreads();

  // ---- Per-block path identity ------------------------------------------
  const int bid = blockIdx.x;            // ((ti*nrx)+r)*Pn + p
  const int p   = bid % Pn;
  const int r   = (bid / Pn) % nrx;
  const int ti  = bid / (Pn * nrx);

  // ---- Image method (order 2), computed redundantly per thread ----------
  // (all loads are workgroup-uniform -> serviced from cache)
  F3 tx = ld3(txv + 3 * ti);
  F3 rx = ld3(rxv + 3 * r);
  const int c0 = pc[2 * p + 0];
  const int c1 = pc[2 * p + 1];
  F3 mn0 = ld3(normals + 3 * c0);
  F3 mn1 = ld3(normals + 3 * c1);
  const float* ta = s_tri + 9 * c0;
  const float* tb = s_tri + 9 * c1;
  F3 v0a = ld3(ta), e1a = ld3(ta + 3), e2a = ld3(ta + 6);   // mirror 0 (mv0 == v0a)
  F3 v0b = ld3(tb), e1b = ld3(tb + 3), e2b = ld3(tb + 6);   // mirror 1 (mv1 == v0b)

  F3 img0 = sub3(tx,   scl3(mn0, 2.0f * dot3(sub3(tx,   v0a), mn0)));
  F3 img1 = sub3(img0, scl3(mn1, 2.0f * dot3(sub3(img0, v0b), mn1)));

  F3 inc1   = sub3(img1, rx);
  float tS1 = dot3(sub3(v0b, rx), mn1) / dot3(inc1, mn1);   // plain divide, like reference
  F3 p1     = add3(rx, scl3(inc1, tS1));                    // point on mirror 1
  F3 inc0   = sub3(img0, p1);
  float tS0 = dot3(sub3(v0a, p1), mn0) / dot3(inc0, mn0);
  F3 p0     = add3(p1, scl3(inc0, tS0));                    // point on mirror 0

  F3 d0 = sub3(p0, tx), d1 = sub3(p1, p0), d2 = sub3(rx, p1);

  // ---- Validity checks ---------------------------------------------------
  float tdum; bool h0, h1;
  mt(tx, d0, v0a, e1a, e2a, tdum, h0);     // segment 0 vs own mirror triangle 0
  mt(p0, d1, v0b, e1b, e2b, tdum, h1);     // segment 1 vs own mirror triangle 1
  bool inside = h0 && h1;

  bool validr =
      (dot3(sub3(tx, v0a), mn0) * dot3(sub3(p1, v0a), mn0) >= 0.0f) &&
      (dot3(sub3(p0, v0b), mn1) * dot3(sub3(rx, v0b), mn1) >= 0.0f);

  bool tooSmall = (dot3(d0, d0) < MIN_LENF) ||
                  (dot3(d1, d1) < MIN_LENF) ||
                  (dot3(d2, d2) < MIN_LENF);

  bool fin = isfinite(tx.x) && isfinite(tx.y) && isfinite(tx.z) &&
             isfinite(p0.x) && isfinite(p0.y) && isfinite(p0.z) &&
             isfinite(p1.x) && isfinite(p1.y) && isfinite(p1.z) &&
             isfinite(rx.x) && isfinite(rx.y) && isfinite(rx.z);

  // ---- Occlusion: 3 segments vs ALL triangles, strip-mined over threads --
  bool myBlocked = false;
  for (int t = tid; t < T; t += BLOCK) {
    const float* tp = s_tri + 9 * t;
    F3 v0 = ld3(tp), e1 = ld3(tp + 3), e2 = ld3(tp + 6);
    float th; bool hh;
    mt(tx, d0, v0, e1, e2, th, hh);
    myBlocked |= (hh && (th > HIT_TOLF) && (th < 1.0f - HIT_TOLF));
    mt(p0, d1, v0, e1, e2, th, hh);
    myBlocked |= (hh && (th > HIT_TOLF) && (th < 1.0f - HIT_TOLF));
    mt(p1, d2, v0, e1, e2, th, hh);
    myBlocked |= (hh && (th > HIT_TOLF) && (th < 1.0f - HIT_TOLF));
  }
  if (myBlocked) atomicOr(&s_blocked, 1);
  __syncthreads();

  // ---- Emit outputs ------------------------------------------------------
  if (tid == 0) {
    bool blocked = (s_blocked != 0);
    bool m = inside && validr && !blocked && !tooSmall && fin;

    float q[12] = { tx.x, tx.y, tx.z,  p0.x, p0.y, p0.z,
                    p1.x, p1.y, p1.z,  rx.x, rx.y, rx.z };
    float* fp = outPaths + (size_t)bid * 12;
#pragma unroll
    for (int i = 0; i < 12; ++i) fp[i] = fin ? q[i] : 0.0f;

    outMask[bid] = m ? 1.0f : 0.0f;

    float* ob = outObj + (size_t)bid * 4;
    ob[0] = (float)ti;
    ob[1] = (float)c0;
    ob[2] = (float)c1;
    ob[3] = (float)r;
  }
}

extern "C" void kernel_launch(void* const* d_in, const int* in_sizes, int n_in,
                              void* d_out, int out_size, void* d_ws, size_t ws_size,
                              hipStream_t stream) {
  (void)n_in; (void)out_size;
  const float* verts   = (const float*)d_in[0];   // (V,3) f32
  const float* normals = (const float*)d_in[1];   // (T,3) f32
  const float* txv     = (const float*)d_in[2];   // (ntx,3) f32
  const float* rxv     = (const float*)d_in[3];   // (nrx,3) f32
  const int*   tris    = (const int*)d_in[4];     // (T,3) i32
  const int*   pc      = (const int*)d_in[5];     // (P,2) i32  (order == 2)

  const int T   = in_sizes[1] / 3;
  const int ntx = in_sizes[2] / 3;
  const int nrx = in_sizes[3] / 3;
  const int Pn  = in_sizes[5] / 2;

  float* out = (float*)d_out;
  const size_t nPath = (size_t)ntx * (size_t)nrx * (size_t)Pn;
  float* outPaths = out;                 // nPath * 12
  float* outMask  = out + nPath * 12;    // nPath   (bool -> 0.0/1.0)
  float* outObj   = out + nPath * 13;    // nPath * 4 (int32 -> float)

  float*       triData = (float*)d_ws;
  const size_t wsNeed  = (size_t)T * 9 * sizeof(float);
  const int    useWs   = (d_ws != nullptr && ws_size >= wsNeed) ? 1 : 0;

  if (useWs) {
    tri_prep_kernel<<<(T + 255) / 256, 256, 0, stream>>>(verts, tris, triData, T);
  }
  trace_kernel<<<dim3((unsigned)nPath), dim3(BLOCK), 0, stream>>>(
      verts, normals, txv, rxv, tris, pc, triData, useWs,
      outPaths, outMask, outObj, T, Pn, nrx);
}